// LogicConv2d_88880053224061
// MI455X (gfx1250) — compile-verified
//
#include <hip/hip_runtime.h>
#include <hip/hip_bf16.h>
#include <stdint.h>

// Problem constants (from reference): B=16, C=3, H=W=64, K=32, S=32, R=5,
// OH=OW=60, P=3600. Tree: 32+16+8+4+2+1 = 63 nodes per filter k.
#define KF 32
#define PP_ 3600
#define OWX 60
#define TILE_POS 128
#define NTILES 29              // ceil(3600/128)
#define SLICE_ROWS 8           // oh span per tile (<=4) + receptive field (5) <= 8
#define XSLICE (3 * SLICE_ROWS * 64)   // 1536 floats staged per block
#define NNODES 63
#define NROWS (NNODES * KF)    // 2016 (node,filter) coefficient rows

typedef __attribute__((ext_vector_type(2))) float v2f;
typedef __attribute__((ext_vector_type(8))) float v8f;
typedef __attribute__((ext_vector_type(4))) int   v4i;

#define GAS __attribute__((address_space(1)))
#define LAS __attribute__((address_space(3)))

#if __has_builtin(__builtin_amdgcn_global_load_async_to_lds_b128) && \
    __has_builtin(__builtin_amdgcn_s_wait_asynccnt)
#define USE_ASYNC 1
#else
#define USE_ASYNC 0
#endif

// ---------------------------------------------------------------------------
// Kernel 1: coefficient build.  For each of 2016 rows r=(node_g*32+k):
//   p = softmax(logits[r])  (16 ops);  c[t] = sum_o p[o] * ((o>>t)&1), t=0..3
// The contraction p(16 rows x 16) @ LUT(16x4) is done with
// V_WMMA_F32_16X16X4_F32, K-chunked 4x.  Output stored in delta form:
//   {c0, c2-c0, c1-c0, c0-c1-c2+c3}  so a combine is 1 mul + 3 fma.
// One wave (32 threads) per 16-row tile; EXEC is all-ones around the WMMAs.
// ---------------------------------------------------------------------------
__global__ __launch_bounds__(32) void build_coeffs(
    const float* __restrict__ w0, const float* __restrict__ w1,
    const float* __restrict__ w2, const float* __restrict__ w3,
    const float* __restrict__ w4, const float* __restrict__ w5,
    float* __restrict__ ctab /* [2016*4] delta coeffs */)
{
    __shared__ __attribute__((aligned(16))) float probs[16][16];
    __shared__ __attribute__((aligned(16))) float rawc[16][16];
    const int lane = threadIdx.x;
    const int tile = blockIdx.x;              // 0..125

    if (lane < 16) {
        const int r = tile * 16 + lane;       // 0..2015
        const int g = r >> 5;                 // global node id 0..62
        const int k = r & 31;
        const float* wp; int node;
        if      (g < 32) { wp = w0; node = g;      }
        else if (g < 48) { wp = w1; node = g - 32; }
        else if (g < 56) { wp = w2; node = g - 48; }
        else if (g < 60) { wp = w3; node = g - 56; }
        else if (g < 62) { wp = w4; node = g - 60; }
        else             { wp = w5; node = 0;      }
        const float* lg = wp + (node * KF + k) * 16;
        float mx = lg[0];
        #pragma unroll
        for (int j = 1; j < 16; ++j) mx = fmaxf(mx, lg[j]);
        float e[16]; float s = 0.0f;
        #pragma unroll
        for (int j = 0; j < 16; ++j) { e[j] = __expf(lg[j] - mx); s += e[j]; }
        const float inv = 1.0f / s;
        #pragma unroll
        for (int j = 0; j < 16; ++j) probs[lane][j] = e[j] * inv;
    }
    __syncthreads();

    const int m  = lane & 15;                 // A row / B col index
    const int hi = (lane >= 16) ? 1 : 0;

#if __has_builtin(__builtin_amdgcn_wmma_f32_16x16x4_f32)
    // D(16x16) += A(16x4) * B(4x16), 4 K-chunks cover the 16 ops.
    // A layout (32b, 16x4): VGPR0 lanes0-15 K=0, lanes16-31 K=2; VGPR1: K=1/K=3.
    // B row k, col n: LUT[o][n] = (o>>n)&1  (zero for n>=4 automatically).
    v8f acc = {};
    #pragma unroll
    for (int chunk = 0; chunk < 4; ++chunk) {
        const int o0 = chunk * 4 + (hi ? 2 : 0);
        const int o1 = chunk * 4 + (hi ? 3 : 1);
        v2f a, b;
        a.x = probs[m][o0];
        a.y = probs[m][o1];
        b.x = (float)((o0 >> m) & 1);
        b.y = (float)((o1 >> m) & 1);
        acc = __builtin_amdgcn_wmma_f32_16x16x4_f32(
            false, a, false, b, (short)0, acc, false, false);
    }
    // C/D layout: lane L, VGPR j -> M = j + (L>=16 ? 8 : 0), N = L&15.
    if (m < 4) {
        #pragma unroll
        for (int j = 0; j < 8; ++j) rawc[hi * 8 + j][m] = acc[j];
    }
#else
    if (lane < 16) {
        #pragma unroll
        for (int t = 0; t < 4; ++t) {
            float s = 0.0f;
            #pragma unroll
            for (int o = 0; o < 16; ++o) s += probs[lane][o] * (float)((o >> t) & 1);
            rawc[lane][t] = s;
        }
    }
#endif
    __syncthreads();

    if (lane < 16) {
        const float c0 = rawc[lane][0], c1 = rawc[lane][1];
        const float c2 = rawc[lane][2], c3 = rawc[lane][3];
        const int r = tile * 16 + lane;
        float4 d;
        d.x = c0; d.y = c2 - c0; d.z = c1 - c0; d.w = c0 - c1 - c2 + c3;
        *(float4*)(ctab + r * 4) = d;
    }
}

// ---------------------------------------------------------------------------
// Kernel 1b: fuse (c,h,w) leaf indices into one LDS-slice offset each.
// Slice layout: channel stride = SLICE_ROWS*64, row stride = 64.
// ---------------------------------------------------------------------------
__global__ __launch_bounds__(256) void build_offs(
    const int* __restrict__ h, const int* __restrict__ w,
    const int* __restrict__ c, int* __restrict__ offs)
{
    const int i = blockIdx.x * 256 + threadIdx.x;   // 0..2047  (s*1024 + k*32 + n)
    if (i < 2048) offs[i] = c[i] * (SLICE_ROWS * 64) + h[i] * 64 + w[i];
}

// ---------------------------------------------------------------------------
// Main kernel: one block = (image b, tile of 128 output positions).
// Stages coeffs + offsets + the 8 needed x-rows into LDS (async-to-LDS path),
// then each thread evaluates the full 63-node tree for (k, pos) pairs.
// k is wave-uniform -> coeff ds_load_b128 broadcasts; gathers are stride-1.
// ---------------------------------------------------------------------------
__global__ __launch_bounds__(256) void logic_conv_main(
    const float* __restrict__ x,      // (16,3,64,64)
    const float* __restrict__ ctab,   // [2016*4] delta coeffs (ws)
    const int*   __restrict__ offs,   // [2048] fused leaf offsets (ws)
    float* __restrict__ out)          // (16,32,60,60)
{
    __shared__ __attribute__((aligned(16))) float sC[NROWS * 4];   // 32256 B
    __shared__ __attribute__((aligned(16))) int   sOff[2048];      //  8192 B
    __shared__ __attribute__((aligned(16))) float sX[XSLICE];      //  6144 B

    const int t    = threadIdx.x;
    const int bimg = blockIdx.x / NTILES;
    const int tile = blockIdx.x % NTILES;
    const int p0   = tile * TILE_POS;
    int r0 = p0 / OWX;                      // first oh in tile
    if (r0 > 64 - SLICE_ROWS) r0 = 64 - SLICE_ROWS;   // keep slice in-bounds

    // Stage delta coefficients: 2016 x float4.
    {
        const float4* src = (const float4*)ctab;
        float4* dst = (float4*)sC;
        for (int i = t; i < NROWS; i += 256) dst[i] = src[i];
    }
    // Stage fused offsets: 512 x int4.
    {
        const int4* src = (const int4*)offs;
        int4* dst = (int4*)sOff;
        for (int i = t; i < 512; i += 256) dst[i] = src[i];
    }
    // Stage x slice: 3 channels x 8 rows x 64 floats = 384 x 16B transfers.
    {
        const float* xb = x + (size_t)bimg * 3 * 64 * 64;
        for (int i = t; i < 384; i += 256) {
            const int c   = i >> 7;
            const int rem = i & 127;
            const int rl  = rem >> 4;
            const int w4  = rem & 15;
            const float* g = xb + c * 4096 + (r0 + rl) * 64 + w4 * 4;
#if USE_ASYNC
            __builtin_amdgcn_global_load_async_to_lds_b128(
                (GAS v4i*)g, (LAS v4i*)&sX[i * 4], 0, 0);
#else
            *(float4*)&sX[i * 4] = *(const float4*)g;
#endif
        }
    }
#if USE_ASYNC
    __builtin_amdgcn_s_wait_asynccnt(0);
#endif
    __syncthreads();

    const int GB[5] = {32, 48, 56, 60, 62};   // level base node ids (lvl 1..5)

    for (int pp = t; pp < KF * TILE_POS; pp += 256) {
        const int k   = pp >> 7;              // wave-uniform (128 % 32 == 0)
        const int pos = p0 + (pp & 127);
        if (pos >= PP_) continue;
        const int oh   = pos / OWX;
        const int ow   = pos - oh * OWX;
        const int base = (oh - r0) * 64 + ow;
        const int ko   = k << 5;

        float y[32];
        #pragma unroll
        for (int n = 0; n < 32; ++n) {
            const float a = sX[sOff[ko + n] + base];
            const float b = sX[sOff[1024 + ko + n] + base];
            const float4 d = *(const float4*)&sC[((n << 5) + k) << 2];
            y[n] = fmaf(a * b, d.w, fmaf(b, d.z, fmaf(a, d.y, d.x)));
        }
        #pragma unroll
        for (int lvl = 0; lvl < 5; ++lvl) {
            const int nn = 16 >> lvl;
            const int gb = GB[lvl];
            #pragma unroll
            for (int n = 0; n < nn; ++n) {
                const float a = y[2 * n], b = y[2 * n + 1];
                const float4 d = *(const float4*)&sC[(((gb + n) << 5) + k) << 2];
                y[n] = fmaf(a * b, d.w, fmaf(b, d.z, fmaf(a, d.y, d.x)));
            }
        }
        out[((size_t)bimg * KF + k) * PP_ + pos] = y[0];
    }
}

// ---------------------------------------------------------------------------
extern "C" void kernel_launch(void* const* d_in, const int* in_sizes, int n_in,
                              void* d_out, int out_size, void* d_ws, size_t ws_size,
                              hipStream_t stream) {
    const float* x     = (const float*)d_in[0];
    const int*   h_idx = (const int*)d_in[1];
    const int*   w_idx = (const int*)d_in[2];
    const int*   c_idx = (const int*)d_in[3];
    const float* w0 = (const float*)d_in[4];
    const float* w1 = (const float*)d_in[5];
    const float* w2 = (const float*)d_in[6];
    const float* w3 = (const float*)d_in[7];
    const float* w4 = (const float*)d_in[8];
    const float* w5 = (const float*)d_in[9];

    float* ctab = (float*)d_ws;                                      // 2016*4 floats
    int*   offs = (int*)((char*)d_ws + NROWS * 4 * sizeof(float));   // 2048 ints
    float* out  = (float*)d_out;

    build_coeffs<<<126, 32, 0, stream>>>(w0, w1, w2, w3, w4, w5, ctab);
    build_offs<<<8, 256, 0, stream>>>(h_idx, w_idx, c_idx, offs);
    logic_conv_main<<<16 * NTILES, 256, 0, stream>>>(x, ctab, offs, out);
}